// VertexAlign_19069654794324
// MI455X (gfx1250) — compile-verified
//
#include <hip/hip_runtime.h>
#include <hip/hip_bf16.h>

typedef float f32x4 __attribute__((ext_vector_type(4)));

// One pyramid level. S is the spatial size (56/28/14/7), LVL the level index.
// Called with wave-uniform LVL (64 threads per level in a 256-thread block),
// so the switch in the caller is non-divergent per wave32.
template <int S, int LVL>
__device__ __forceinline__ void sample_level(const float* __restrict__ f,
                                             int b, int c0, float hh, float ww,
                                             float* __restrict__ outrow)
{
    // S/224 is an exact power of two (1/4, 1/8, 1/16, 1/32):
    // multiplying by it is bit-identical to the reference's division by 224/S.
    constexpr float inv = (float)S / 224.0f;
    const float x = ww * inv;
    const float y = hh * inv;

    const int x1 = (int)floorf(x);                 // == trunc(x) for x >= 0
    const int y1 = (int)floorf(y);
    const int x2 = min((int)ceilf(x), S - 1);
    const int y2 = min((int)ceilf(y), S - 1);

    // Reference's integer weights: w12 = w21 = w22 = 0 identically,
    // w11 = (x2-x1)*(y2-y1) in {0,1}.
    const int wi = (x2 - x1) * (y2 - y1);
    const float wgt = (float)wi;

    f32x4 v = {0.0f, 0.0f, 0.0f, 0.0f};
    if (wi != 0) {                                  // wave-uniform branch
        // feat layout (B, 256, S, S); gather one texel across 4 channels.
        const long hw   = (long)(S * S);
        const long base = ((long)(b * 256 + c0)) * hw + (long)(x1 * S + y1);
        v.x = f[base];
        v.y = f[base + hw];
        v.z = f[base + 2 * hw];
        v.w = f[base + 3 * hw];
        v *= wgt;                                   // wgt == 1 here; keeps exact semantics
    }

    // 512 MB of write-once output: non-temporal 128-bit store (TH=NT) so the
    // streaming output does not evict the L2-resident feature pyramids.
    __builtin_nontemporal_store(v, (f32x4*)(outrow + LVL * 256 + c0));
}

__global__ __launch_bounds__(256)
void vertex_align_kernel(const float* __restrict__ f0,
                         const float* __restrict__ f1,
                         const float* __restrict__ f2,
                         const float* __restrict__ f3,
                         const float* __restrict__ pos,
                         float*       __restrict__ out,
                         int V)                      // verts per mesh (host-derived)
{
    const int b = blockIdx.y;                 // batch index — no integer division
    const int n = b * V + blockIdx.x;         // global vertex index
    const int t = threadIdx.x;                // 0..255

    const float px = pos[3 * n + 0];
    const float py = pos[3 * n + 1];
    const float pz = pos[3 * n + 2];

    float h = 248.0f * (py / pz)    + 111.5f;
    float w = 248.0f * (px / (-pz)) + 111.5f;
    h = fminf(fmaxf(h, 0.0f), 223.0f);
    w = fminf(fmaxf(w, 0.0f), 223.0f);

    float* outrow = out + (long)n * 1024;

    const int level = t >> 6;        // 64 threads per level -> 2 waves per level
    const int c0    = (t & 63) << 2; // 4 consecutive channels per thread

    switch (level) {
        case 0:  sample_level<56, 0>(f0, b, c0, h, w, outrow); break;
        case 1:  sample_level<28, 1>(f1, b, c0, h, w, outrow); break;
        case 2:  sample_level<14, 2>(f2, b, c0, h, w, outrow); break;
        default: sample_level< 7, 3>(f3, b, c0, h, w, outrow); break;
    }
}

extern "C" void kernel_launch(void* const* d_in, const int* in_sizes, int n_in,
                              void* d_out, int out_size, void* d_ws, size_t ws_size,
                              hipStream_t stream)
{
    const float* f0  = (const float*)d_in[0];  // (B,256,56,56)
    const float* f1  = (const float*)d_in[1];  // (B,256,28,28)
    const float* f2  = (const float*)d_in[2];  // (B,256,14,14)
    const float* f3  = (const float*)d_in[3];  // (B,256,7,7)
    const float* pos = (const float*)d_in[4];  // (N,3)

    const int N = in_sizes[4] / 3;              // total vertices (B*V)
    const int B = in_sizes[0] / (256 * 56 * 56); // batch count from feat0 shape
    const int V = N / B;                         // verts per mesh (setup: N = B*V)

    dim3 grid(V, B, 1);
    vertex_align_kernel<<<grid, 256, 0, stream>>>(f0, f1, f2, f3, pos,
                                                  (float*)d_out, V);
}